// LSMDecoder_42666205118482
// MI455X (gfx1250) — compile-verified
//
#include <hip/hip_runtime.h>
#include <hip/hip_bf16.h>

// ---------------------------------------------------------------------------
// LSM decoder, fused:
//   phase 0: column softmax of the three 128x32 transforms      (3 blocks)
//   phase 1: per-row projections -> A (N,64), B (N,64), C (N)   (1024 blocks)
//   phase 2: x = C[i] + NUM_EDGES*A@B^T (N x N x 64 fp32 WMMA GEMM), sigmoid epilogue
// ---------------------------------------------------------------------------

#define NROWS      8192
#define IN_DIM     128
#define LAT        32
#define KDIM       64            // 2*LAT
#define NUM_EDGES  100000.0f
#define SMALLV     1e-16f

typedef float v2f __attribute__((ext_vector_type(2)));
typedef float v8f __attribute__((ext_vector_type(8)));

// ---------------- phase 0: softmax over axis 0 of a (128,32) matrix --------
__global__ __launch_bounds__(32)
void lsm_softmax_kernel(const float* __restrict__ t0,
                        const float* __restrict__ t1,
                        const float* __restrict__ t2,
                        float* __restrict__ ws) {
    const float* src = (blockIdx.x == 0) ? t0 : (blockIdx.x == 1) ? t1 : t2;
    float* dst = ws + (size_t)blockIdx.x * (IN_DIM * LAT);
    const int j = threadIdx.x;                    // one column per lane (wave32)
    float mx = -3.0e38f;
    for (int r = 0; r < IN_DIM; ++r) mx = fmaxf(mx, src[r * LAT + j]);
    float sum = 0.0f;
    for (int r = 0; r < IN_DIM; ++r) sum += __expf(src[r * LAT + j] - mx);
    const float inv = 1.0f / sum;
    for (int r = 0; r < IN_DIM; ++r)
        dst[r * LAT + j] = __expf(src[r * LAT + j] - mx) * inv;
}

// ---------------- phase 1: per-row decode + GEMM operand construction ------
// One wave per row i; lane d handles latent dim d (wave32 == LAT).
__global__ __launch_bounds__(256)
void lsm_rows_kernel(const float* __restrict__ z1,
                     const float* __restrict__ gamma,
                     const float* __restrict__ z2,
                     const float* __restrict__ delta,
                     const float* __restrict__ wst,      // softmaxed transforms
                     const float* __restrict__ bias_p,
                     const float* __restrict__ wg_p,
                     const float* __restrict__ wd_p,
                     float* __restrict__ Amat,
                     float* __restrict__ Bmat,
                     float* __restrict__ Cvec,
                     float* __restrict__ o_z1,
                     float* __restrict__ o_z2,
                     float* __restrict__ o_g,
                     float* __restrict__ o_d) {
    const int wave = threadIdx.x >> 5;
    const int lane = threadIdx.x & 31;
    const int i = blockIdx.x * 8 + wave;

    const float* tz = wst;
    const float* tg = wst + IN_DIM * LAT;
    const float* td = wst + 2 * IN_DIM * LAT;
    const float* z1r = z1    + (size_t)i * IN_DIM;
    const float* z2r = z2    + (size_t)i * IN_DIM;
    const float* gr  = gamma + (size_t)i * IN_DIM;
    const float* dr  = delta + (size_t)i * IN_DIM;

    float x1 = 0.f, x2 = 0.f, gd = 0.f, dd = 0.f;
    for (int k = 0; k < IN_DIM; ++k) {
        const float tzk = tz[k * LAT + lane];
        x1 = fmaf(z1r[k], tzk,              x1);
        x2 = fmaf(z2r[k], tzk,              x2);
        gd = fmaf(gr[k],  tg[k * LAT + lane], gd);
        dd = fmaf(dr[k],  td[k * LAT + lane], dd);
    }

    const float wg = *wg_p, wd = *wd_p, bias = *bias_p;
    const float s = wg * (gd + SMALLV) + wd * (dd + SMALLV);

    Amat[(size_t)i * KDIM + lane]       = 2.0f * s * x1;
    Amat[(size_t)i * KDIM + LAT + lane] = -s;
    Bmat[(size_t)i * KDIM + lane]       = x2;
    Bmat[(size_t)i * KDIM + LAT + lane] = x2 * x2;

    o_z1[(size_t)i * LAT + lane] = x1;
    o_z2[(size_t)i * LAT + lane] = x2;
    o_g [(size_t)i * LAT + lane] = gd;
    o_d [(size_t)i * LAT + lane] = dd;

    float p = s * x1 * x1;                         // wave32 reduction
    for (int off = 16; off > 0; off >>= 1) p += __shfl_xor(p, off, 32);
    if (lane == 0) Cvec[i] = bias - NUM_EDGES * p;
}

// ---------------- phase 2: 128x128-tile GEMM (K=64) + sigmoid --------------
// 8 waves; wave (wr=w&3, wc=w>>2) owns a 32x64 block = 2x4 WMMA tiles.
// fp32 WMMA 16x16x4: A frag lane layout (m=lane%16, k pair = 2*(lane/16)),
// B frag mirrors it with n=lane%16. C/D: vgpr r -> row r+8*(lane/16).
#define LDS_STRIDE 68   // multiple of 4 (aligned float4 stores), conflict-free

__global__ __launch_bounds__(256)
void lsm_gemm_sigmoid_kernel(const float* __restrict__ Amat,
                             const float* __restrict__ Bmat,
                             const float* __restrict__ Cvec,
                             float* __restrict__ out) {
    __shared__ float As[128][LDS_STRIDE];
    __shared__ float Bs[128][LDS_STRIDE];
    __shared__ float Cs[128];

    const int tid = threadIdx.x;
    const int i0 = blockIdx.y * 128;
    const int j0 = blockIdx.x * 128;

    // stage 128x64 A and B panels (float4, fully coalesced)
    for (int idx = tid; idx < 128 * (KDIM / 4); idx += 256) {
        const int r  = idx >> 4;
        const int c4 = (idx & 15) << 2;
        const float4 av = *(const float4*)(Amat + (size_t)(i0 + r) * KDIM + c4);
        const float4 bv = *(const float4*)(Bmat + (size_t)(j0 + r) * KDIM + c4);
        *(float4*)(&As[r][c4]) = av;
        *(float4*)(&Bs[r][c4]) = bv;
    }
    if (tid < 128) Cs[tid] = Cvec[i0 + tid];
    __syncthreads();

    const int wave = tid >> 5;
    const int lane = tid & 31;
    const int m0 = (wave & 3) * 32;     // row offset of this wave's block
    const int n0 = (wave >> 2) * 64;    // col offset of this wave's block
    const int mr = lane & 15;
    const int kk = (lane >> 4) << 1;    // 0 or 2: this lane's k-pair base

    v8f acc[2][4] = {};
    for (int k = 0; k < KDIM; k += 4) {
        v2f a[2], b[4];
        a[0] = *(const v2f*)&As[m0 +      mr][k + kk];
        a[1] = *(const v2f*)&As[m0 + 16 + mr][k + kk];
        b[0] = *(const v2f*)&Bs[n0 +      mr][k + kk];
        b[1] = *(const v2f*)&Bs[n0 + 16 + mr][k + kk];
        b[2] = *(const v2f*)&Bs[n0 + 32 + mr][k + kk];
        b[3] = *(const v2f*)&Bs[n0 + 48 + mr][k + kk];
#pragma unroll
        for (int r = 0; r < 2; ++r)
#pragma unroll
            for (int c = 0; c < 4; ++c)
                acc[r][c] = __builtin_amdgcn_wmma_f32_16x16x4_f32(
                    false, a[r], false, b[c], (short)0, acc[r][c], false, false);
    }

    // epilogue: sigmoid(C[i] + NUM_EDGES * acc), write straight to HBM
    const int rhalf = (lane >> 4) << 3;   // 0 or 8
    const int ncol  = j0 + n0 + (lane & 15);
#pragma unroll
    for (int r = 0; r < 2; ++r) {
#pragma unroll
        for (int vr = 0; vr < 8; ++vr) {
            const int lrow = m0 + 16 * r + rhalf + vr;
            const float cv = Cs[lrow];
            const size_t rowbase = (size_t)(i0 + lrow) * NROWS;
#pragma unroll
            for (int c = 0; c < 4; ++c) {
                const float x  = cv + NUM_EDGES * acc[r][c][vr];
                const float sg = 1.0f / (1.0f + __expf(-x));
                out[rowbase + ncol + 16 * c] = sg;
            }
        }
    }
}

// ---------------------------------------------------------------------------
extern "C" void kernel_launch(void* const* d_in, const int* in_sizes, int n_in,
                              void* d_out, int out_size, void* d_ws, size_t ws_size,
                              hipStream_t stream) {
    const float* z1    = (const float*)d_in[0];
    const float* gamma = (const float*)d_in[1];
    const float* z2    = (const float*)d_in[2];
    const float* delta = (const float*)d_in[3];
    const float* t_z   = (const float*)d_in[4];
    const float* t_g   = (const float*)d_in[5];
    const float* t_d   = (const float*)d_in[6];
    const float* bias  = (const float*)d_in[7];
    const float* w_g   = (const float*)d_in[8];
    const float* w_d   = (const float*)d_in[9];

    // workspace layout (floats): [3*128*32 softmaxed transforms][A][B][C]
    float* ws   = (float*)d_ws;
    float* wst  = ws;
    float* Amat = wst + 3 * IN_DIM * LAT;
    float* Bmat = Amat + (size_t)NROWS * KDIM;
    float* Cvec = Bmat + (size_t)NROWS * KDIM;

    // output layout: [N*N sigmoid][z_dec1][z_dec2][gam_dec][del_dec]
    float* out   = (float*)d_out;
    float* o_z1  = out  + (size_t)NROWS * NROWS;
    float* o_z2  = o_z1 + (size_t)NROWS * LAT;
    float* o_g   = o_z2 + (size_t)NROWS * LAT;
    float* o_d   = o_g  + (size_t)NROWS * LAT;

    lsm_softmax_kernel<<<3, 32, 0, stream>>>(t_z, t_g, t_d, wst);
    lsm_rows_kernel<<<NROWS / 8, 256, 0, stream>>>(z1, gamma, z2, delta, wst,
                                                   bias, w_g, w_d,
                                                   Amat, Bmat, Cvec,
                                                   o_z1, o_z2, o_g, o_d);
    dim3 grid(NROWS / 128, NROWS / 128);
    lsm_gemm_sigmoid_kernel<<<grid, 256, 0, stream>>>(Amat, Bmat, Cvec, out);
}